// GCNLayer_1864015807058
// MI455X (gfx1250) — compile-verified
//
#include <hip/hip_runtime.h>

#define D_IN  256
#define D_OUT 128

typedef __attribute__((ext_vector_type(2))) float v2f;
typedef __attribute__((ext_vector_type(4))) float v4f;
typedef __attribute__((ext_vector_type(8))) float v8f;

// Non-returning device-scope fp32 atomic add: RMW executes at L2 (device
// coherence point), no return data (TH[0]=0 -> STOREcnt). This is the path
// the MI455X L2 rewards for scatter-add into an L2-resident accumulator.
__device__ __forceinline__ void atomic_add_f32_dev(float* p, float v) {
    asm volatile("global_atomic_add_f32 %0, %1, off scope:SCOPE_DEV"
                 :: "v"(p), "v"(v) : "memory");
}

// ---------------------------------------------------------------------------
// Zero the output accumulator (d_out is poisoned by the harness).
// ---------------------------------------------------------------------------
__global__ __launch_bounds__(256) void gcn_zero(v4f* __restrict__ p, int n4) {
    int i = blockIdx.x * blockDim.x + threadIdx.x;
    if (i < n4) p[i] = (v4f){0.f, 0.f, 0.f, 0.f};
}

// ---------------------------------------------------------------------------
// Transpose W[256][128] -> Wt[128][256] so WMMA B-fragments (K-contiguous per
// output column) become contiguous b64 loads.
// ---------------------------------------------------------------------------
__global__ __launch_bounds__(256) void gcn_transpose_w(const float* __restrict__ w,
                                                       float* __restrict__ wt) {
    int idx = blockIdx.x * blockDim.x + threadIdx.x;   // over 256*128
    if (idx < D_IN * D_OUT) {
        int k = idx / D_OUT;
        int n = idx % D_OUT;
        wt[n * D_IN + k] = w[idx];
    }
}

// ---------------------------------------------------------------------------
// support = x @ W using V_WMMA_F32_16X16X4_F32.
// Block = 256 threads = 8 waves. Block computes rows [16b, 16b+16) x all 128
// cols; wave w owns cols [16w, 16w+16). x-tile (16x256 f32, 16KB) is staged in
// LDS; Wt is read from global (128KB, L2/WGP$ resident).
//
// A-frag (16x4 f32, ISA 7.12.2): lane<16 -> (M=lane,   K=k0+0 / k0+1)
//                                lane>=16 -> (M=lane-16, K=k0+2 / k0+3)
// B-frag (4x16) symmetric with M->N. Both are contiguous 8-byte loads.
// C/D: VGPR r -> M = r + (lane>=16 ? 8 : 0), N = lane & 15.
//
// x is streamed once (102MB total) -> nontemporal loads so it does not evict
// the L2-resident support/out working set of the scatter phase.
// ---------------------------------------------------------------------------
__global__ __launch_bounds__(256) void gcn_gemm_wmma(
    const float* __restrict__ x, const float* __restrict__ wt,
    float* __restrict__ support, int n_nodes)
{
    __shared__ float xs[16 * D_IN];
    const int tid  = threadIdx.x;
    const int row0 = blockIdx.x * 16;

    // Cooperative tile load: 1024 float4s across 256 threads (4 each).
    {
        const v4f* xg  = (const v4f*)x;
        v4f*       xls = (v4f*)xs;
#pragma unroll
        for (int i = 0; i < 4; ++i) {
            int f4   = tid + 256 * i;       // 0..1023
            int r    = f4 >> 6;             // 64 float4 per row
            int c    = f4 & 63;
            int grow = row0 + r;
            v4f v = {0.f, 0.f, 0.f, 0.f};
            if (grow < n_nodes)
                v = __builtin_nontemporal_load(&xg[(size_t)grow * (D_IN / 4) + c]);
            xls[f4] = v;
        }
    }
    __syncthreads();

    const int wave    = tid >> 5;
    const int lane    = tid & 31;
    const int laneLow = lane & 15;
    const int khalf   = (lane >> 4) << 1;    // 0 or 2
    const int col0    = wave * 16;

    const float* arow = &xs[laneLow * D_IN + khalf];                       // LDS
    const float* brow = &wt[(size_t)(col0 + laneLow) * D_IN + khalf];      // global

    v8f acc = {};
#pragma unroll 8
    for (int k0 = 0; k0 < D_IN; k0 += 4) {
        v2f a = *(const v2f*)(arow + k0);   // ds_load_b64
        v2f b = *(const v2f*)(brow + k0);   // global_load_b64
        acc = __builtin_amdgcn_wmma_f32_16x16x4_f32(
            /*neg_a=*/false, a, /*neg_b=*/false, b,
            /*c_mod=*/(short)0, acc, /*reuse_a=*/false, /*reuse_b=*/false);
    }

    const int mbase = (lane >> 4) << 3;      // 0 or 8
    float* outp = &support[(size_t)(row0 + mbase) * D_OUT + col0 + laneLow];
    if (row0 + 16 <= n_nodes) {
        // Fast path: tile fully interior (wave-uniform branch, no exec toggling)
#pragma unroll
        for (int r = 0; r < 8; ++r) outp[(size_t)r * D_OUT] = acc[r];
    } else {
#pragma unroll
        for (int r = 0; r < 8; ++r)
            if (row0 + mbase + r < n_nodes) outp[(size_t)r * D_OUT] = acc[r];
    }
}

// ---------------------------------------------------------------------------
// Edge scatter: out[dst] += ew * support[src].
// One wave32 per edge; each lane owns 4 consecutive floats (128/32 = 4) ->
// b128 gather + 4 non-returning device-scope f32 atomic adds. Edges staged 32
// at a time via per-lane NT loads (streamed once) and broadcast with __shfl.
// support & out both fit in the 192MB L2, so this loop is L2-bandwidth bound.
// ---------------------------------------------------------------------------
__global__ __launch_bounds__(256) void gcn_scatter(
    const float* __restrict__ support, const int* __restrict__ esrc,
    const int* __restrict__ edst, const float* __restrict__ ew,
    float* __restrict__ out, int n_edges)
{
    const int lane  = threadIdx.x & 31;
    const int gwave = (int)((blockIdx.x * blockDim.x + threadIdx.x) >> 5);
    const int nwave = (int)((gridDim.x * blockDim.x) >> 5);

    for (int base = gwave * 32; base < n_edges; base += nwave * 32) {
        int   e = base + lane;
        int   s = 0, d = 0;
        float w = 0.f;
        if (e < n_edges) {
            s = __builtin_nontemporal_load(&esrc[e]);
            d = __builtin_nontemporal_load(&edst[e]);
            w = __builtin_nontemporal_load(&ew[e]);
        }
        int cnt = n_edges - base;
        if (cnt > 32) cnt = 32;
        for (int j = 0; j < cnt; ++j) {
            int   sj = __shfl(s, j);
            int   dj = __shfl(d, j);
            float wj = __shfl(w, j);
            const v4f m = *(const v4f*)(support + (size_t)sj * D_OUT + lane * 4);
            float* o = out + (size_t)dj * D_OUT + lane * 4;
            atomic_add_f32_dev(o + 0, m.x * wj);
            atomic_add_f32_dev(o + 1, m.y * wj);
            atomic_add_f32_dev(o + 2, m.z * wj);
            atomic_add_f32_dev(o + 3, m.w * wj);
        }
    }
}

// ---------------------------------------------------------------------------
// Fused epilogue: out = relu(out + bias)
// ---------------------------------------------------------------------------
__global__ __launch_bounds__(256) void gcn_epilogue(float* __restrict__ out,
                                                    const float* __restrict__ bias,
                                                    int n) {
    int i = blockIdx.x * blockDim.x + threadIdx.x;
    if (i < n) {
        float v = out[i] + bias[i & (D_OUT - 1)];
        out[i] = fmaxf(v, 0.f);
    }
}

// ---------------------------------------------------------------------------
// Launch. Inputs: x, edge_src, edge_dst, edge_weight, weight, bias.
// Workspace: Wt (128KB) followed by support (n_nodes*128 f32).
// ---------------------------------------------------------------------------
extern "C" void kernel_launch(void* const* d_in, const int* in_sizes, int n_in,
                              void* d_out, int out_size, void* d_ws, size_t ws_size,
                              hipStream_t stream) {
    const float* x    = (const float*)d_in[0];
    const int*   esrc = (const int*)d_in[1];
    const int*   edst = (const int*)d_in[2];
    const float* ew   = (const float*)d_in[3];
    const float* w    = (const float*)d_in[4];
    const float* bias = (const float*)d_in[5];
    float*       out  = (float*)d_out;

    const int n_nodes = in_sizes[0] / D_IN;
    const int n_edges = in_sizes[1];

    float* wt      = (float*)d_ws;
    float* support = wt + (size_t)D_IN * D_OUT;   // 128KB offset, 16B aligned

    // 1) zero accumulator
    int n4 = out_size / 4;
    gcn_zero<<<(n4 + 255) / 256, 256, 0, stream>>>((v4f*)out, n4);

    // 2) transpose W for K-contiguous B fragments
    gcn_transpose_w<<<(D_IN * D_OUT + 255) / 256, 256, 0, stream>>>(w, wt);

    // 3) dense GEMM via f32 WMMA
    gcn_gemm_wmma<<<(n_nodes + 15) / 16, 256, 0, stream>>>(x, wt, support, n_nodes);

    // 4) gather-scale-scatter over edges (256 edges per block per pass)
    int sblocks = (n_edges + 255) / 256;
    gcn_scatter<<<sblocks, 256, 0, stream>>>(support, esrc, edst, ew, out, n_edges);

    // 5) bias + ReLU
    gcn_epilogue<<<(out_size + 255) / 256, 256, 0, stream>>>(out, bias, out_size);
}